// HybridQuanvolutionClassifier_65481071401702
// MI455X (gfx1250) — compile-verified
//
#include <hip/hip_runtime.h>

#define NP   196
#define NPAD 208          // 13 * 16
#define NT   13           // 16-wide tiles covering 196 (+12 zero pad rows)

typedef __attribute__((ext_vector_type(2))) float v2f;
typedef __attribute__((ext_vector_type(8))) float v8f;

__global__ __launch_bounds__(256) void quanv_graph_kernel(
    const float* __restrict__ x,     // [B, 784]  (B,1,28,28)
    const float* __restrict__ phi,   // [B, 196, 4]
    const float* __restrict__ Wlin,  // [10, 784]
    const float* __restrict__ blin,  // [10]
    float* __restrict__ out)         // [B, 10]
{
    __shared__ float nrmL[NPAD * 4];   // normalized states (padded rows zero)
    __shared__ float eL[NPAD * 4];     // expvals (padded rows zero)
    __shared__ float featsL[NP * 4];   // smoothed features
    __shared__ float logitsL[10];

    const int b    = blockIdx.x;
    const int tid  = threadIdx.x;
    const int lane = tid & 31;
    const int wave = tid >> 5;

    // ---------------- Phase 0: patches -> expvals (cumprod) + normalized ----
    if (tid < NP) {
        const int hp = tid / 14, wp = tid % 14;
        const float* xb = x + b * 784;
        const float* pb = phi + (b * NP + tid) * 4;
        float e[4];
        float run = 1.0f;
        #pragma unroll
        for (int k = 0; k < 4; ++k) {
            const int kh = k >> 1, kw = k & 1;
            const float th = xb[(hp * 2 + kh) * 28 + (wp * 2 + kw)];
            const float z  = __cosf(th) * __cosf(pb[k]);
            run *= z;
            e[k] = run;
        }
        const float nn  = sqrtf(e[0]*e[0] + e[1]*e[1] + e[2]*e[2] + e[3]*e[3]) + 1e-12f;
        const float inv = 1.0f / nn;
        #pragma unroll
        for (int k = 0; k < 4; ++k) {
            eL[tid * 4 + k]   = e[k];
            nrmL[tid * 4 + k] = e[k] * inv;
        }
    } else if (tid < NPAD) {
        #pragma unroll
        for (int k = 0; k < 4; ++k) {
            eL[tid * 4 + k]   = 0.0f;
            nrmL[tid * 4 + k] = 0.0f;
        }
    }
    if (tid < 10) logitsL[tid] = blin[tid];
    __syncthreads();

    // ---------------- Phase 1: Gram via WMMA f32 16x16x4, threshold, smooth --
    // C layout (16x16 f32): VGPR r, lanes 0-15 -> row r, col=lane;
    //                               lanes 16-31 -> row r+8, col=lane-16.
    // A (16x4) / B (4x16) f32 per-lane loads coincide when B = A^T (Gram).
    const int hlf = lane >> 4;   // 0: K=0,1 / rows r ; 1: K=2,3 / rows r+8
    const int l15 = lane & 15;

    for (int rt = wave; rt < NT; rt += 8) {          // wave-uniform
        const int rowBase = rt * 16;
        v2f a;
        a.x = nrmL[(rowBase + l15) * 4 + 2 * hlf + 0];
        a.y = nrmL[(rowBase + l15) * 4 + 2 * hlf + 1];

        float degAcc[8];
        float sAcc[8][4];
        #pragma unroll
        for (int r = 0; r < 8; ++r) {
            degAcc[r] = 0.0f;
            sAcc[r][0] = sAcc[r][1] = sAcc[r][2] = sAcc[r][3] = 0.0f;
        }

        for (int ct = 0; ct < NT; ++ct) {            // wave-uniform
            const int colBase = ct * 16;
            v2f bm;
            bm.x = nrmL[(colBase + l15) * 4 + 2 * hlf + 0];
            bm.y = nrmL[(colBase + l15) * 4 + 2 * hlf + 1];

            v8f c = {};
            c = __builtin_amdgcn_wmma_f32_16x16x4_f32(
                    false, a, false, bm, (short)0, c, false, false);

            const int gcol = colBase + l15;
            const float ec0 = eL[gcol * 4 + 0];
            const float ec1 = eL[gcol * 4 + 1];
            const float ec2 = eL[gcol * 4 + 2];
            const float ec3 = eL[gcol * 4 + 3];

            #pragma unroll
            for (int r = 0; r < 8; ++r) {
                const float g   = c[r];
                const float fid = g * g;
                float w = (fid >= 0.8f) ? 1.0f : ((fid >= 0.5f) ? 0.5f : 0.0f);
                const int grow = rowBase + r + 8 * hlf;
                if (grow == gcol) w = 0.0f;          // no self-edges
                degAcc[r]  += w;
                sAcc[r][0] += w * ec0;
                sAcc[r][1] += w * ec1;
                sAcc[r][2] += w * ec2;
                sAcc[r][3] += w * ec3;
            }
        }

        // Butterfly reduce over the 16-lane column groups (masks < 16 stay
        // inside each half-wave, matching the C-matrix row split).
        #pragma unroll
        for (int m = 8; m >= 1; m >>= 1) {
            #pragma unroll
            for (int r = 0; r < 8; ++r) {
                degAcc[r]  += __shfl_xor(degAcc[r],  m);
                sAcc[r][0] += __shfl_xor(sAcc[r][0], m);
                sAcc[r][1] += __shfl_xor(sAcc[r][1], m);
                sAcc[r][2] += __shfl_xor(sAcc[r][2], m);
                sAcc[r][3] += __shfl_xor(sAcc[r][3], m);
            }
        }

        if (l15 == 0) {                              // lanes 0 and 16 write
            #pragma unroll
            for (int r = 0; r < 8; ++r) {
                const int grow = rowBase + r + 8 * hlf;
                if (grow < NP) {
                    const float scale = 1.0f + degAcc[r];
                    #pragma unroll
                    for (int d = 0; d < 4; ++d)
                        featsL[grow * 4 + d] =
                            eL[grow * 4 + d] * scale - sAcc[r][d];
                }
            }
        }
    }
    __syncthreads();

    // ---------------- Phase 2: 784 -> 10 linear + log_softmax ---------------
    float acc[10];
    #pragma unroll
    for (int j = 0; j < 10; ++j) acc[j] = 0.0f;
    for (int i = tid; i < NP * 4; i += 256) {
        const float f = featsL[i];
        #pragma unroll
        for (int j = 0; j < 10; ++j) acc[j] += f * Wlin[j * (NP * 4) + i];
    }
    #pragma unroll
    for (int j = 0; j < 10; ++j) {
        float v = acc[j];
        #pragma unroll
        for (int m = 16; m >= 1; m >>= 1) v += __shfl_xor(v, m);
        if (lane == 0) atomicAdd(&logitsL[j], v);    // ds_add_f32
    }
    __syncthreads();

    if (tid == 0) {
        float l[10];
        float mx = -3.402823466e+38f;
        #pragma unroll
        for (int j = 0; j < 10; ++j) { l[j] = logitsL[j]; mx = fmaxf(mx, l[j]); }
        float s = 0.0f;
        #pragma unroll
        for (int j = 0; j < 10; ++j) s += expf(l[j] - mx);
        const float lg = logf(s) + mx;
        #pragma unroll
        for (int j = 0; j < 10; ++j) out[b * 10 + j] = l[j] - lg;
    }
}

extern "C" void kernel_launch(void* const* d_in, const int* in_sizes, int n_in,
                              void* d_out, int out_size, void* d_ws, size_t ws_size,
                              hipStream_t stream) {
    const float* x    = (const float*)d_in[0];   // [B,1,28,28]
    const float* phi  = (const float*)d_in[1];   // [B,196,4]
    const float* Wlin = (const float*)d_in[2];   // [10,784]
    const float* blin = (const float*)d_in[3];   // [10]
    float* out = (float*)d_out;                  // [B,10]

    const int B = in_sizes[0] / 784;
    (void)n_in; (void)out_size; (void)d_ws; (void)ws_size;
    quanv_graph_kernel<<<dim3(B), dim3(256), 0, stream>>>(x, phi, Wlin, blin, out);
}